// LearnableMovingAvg_83073257439477
// MI455X (gfx1250) — compile-verified
//
#include <hip/hip_runtime.h>
#include <cstdint>

// Problem constants (fixed by the reference setup).
#define B_      64
#define L_      4096
#define C_      64
#define TILE_L  64
#define HALO_F  6          // k=12 -> offsets -6..+5
#define HALO_B  5
#define MAX_ROWS (TILE_L + HALO_F + HALO_B)   // 75 rows incl. halo

typedef unsigned int u32x4 __attribute__((ext_vector_type(4)));
typedef int          i32x4 __attribute__((ext_vector_type(4)));
typedef int          i32x8 __attribute__((ext_vector_type(8)));

__global__ __launch_bounds__(256)
void lma_tdm_kernel(const float* __restrict__ x,
                    const float* __restrict__ mw,   // (5,1)
                    const float* __restrict__ mb,   // (5,)
                    const float* __restrict__ aw,   // (C,)
                    const float* __restrict__ ab,   // (C,)
                    float* __restrict__ out)        // season | trend, each B*L*C
{
    __shared__ float tile[MAX_ROWS * C_];           // 19.2 KB of the 320 KB WGP LDS

    const int bid = blockIdx.x;
    const int b   = bid >> 6;                       // / (L_/TILE_L)
    const int t   = bid & 63;
    const int l0  = t * TILE_L;

    int rs = l0 - HALO_F;          if (rs < 0)       rs = 0;
    int re = l0 + TILE_L - 1 + HALO_B; if (re > L_ - 1) re = L_ - 1;
    const int      row_start = rs;
    const uint32_t nelem     = (uint32_t)(re - rs + 1) * C_;   // <= 4800 elements

    // ---- TDM: one contiguous HBM -> LDS DMA per block (wave 0 issues) ----
    if (threadIdx.x < 32) {
        const uint64_t gaddr = (uint64_t)(uintptr_t)(x + ((size_t)b * L_ + row_start) * C_);
        const uint32_t ldsb  = (uint32_t)(uintptr_t)(void*)tile;  // low 32b = LDS byte offset

        u32x4 g0;
        g0.x = 1u;                                           // count=1 (valid user D#)
        g0.y = ldsb;                                         // lds_addr (bytes)
        g0.z = (uint32_t)gaddr;                              // global_addr[31:0]
        g0.w = (uint32_t)((gaddr >> 32) & 0x1FFFFFFu)        // global_addr[56:32]
             | (2u << 30);                                   // type = 2 ("image")

        i32x8 g1;
        g1[0] = (int)(2u << 16);                             // workgroup_mask=0, data_size=4B
        g1[1] = (int)((nelem & 0xFFFFu) << 16);              // tensor_dim0[15:0]
        g1[2] = (int)(((nelem >> 16) & 0xFFFFu)              // tensor_dim0[31:16]
              |        (1u << 16));                          // tensor_dim1 = 1
        g1[3] = (int)((nelem & 0xFFFFu) << 16);              // tensor_dim1[31:16]=0, tile_dim0=nelem
        g1[4] = 0;                                           // tile_dim1=0 (1-D), tile_dim2=0
        g1[5] = (int)nelem;                                  // tensor_dim0_stride[31:0]
        g1[6] = 0;                                           // stride0[47:32], stride1[15:0]
        g1[7] = 0;                                           // stride1[47:16]

        i32x4 z4 = {0, 0, 0, 0};
#if defined(__clang_major__) && (__clang_major__ >= 23)
        i32x8 z8 = {0, 0, 0, 0, 0, 0, 0, 0};
        __builtin_amdgcn_tensor_load_to_lds(g0, g1, z4, z4, z8, 0);
#else
        __builtin_amdgcn_tensor_load_to_lds(g0, g1, z4, z4, 0);
#endif
        __builtin_amdgcn_s_wait_tensorcnt(0);                // data in LDS before we signal
    }
    __syncthreads();                                         // releases the other 7 waves

    // ---- compute: 12-tap window from LDS, softmax gate, blend, affine ----
    const int c   = threadIdx.x & (C_ - 1);                  // bank == channel: conflict-free
    const int sub = threadIdx.x >> 6;                        // 0..3 row groups

    const float w0 = mw[0], w1 = mw[1], w2 = mw[2], w3 = mw[3], w4 = mw[4];
    const float c0 = mb[0], c1 = mb[1], c2 = mb[2], c3 = mb[3], c4 = mb[4];
    const float awc = aw[c], abc = ab[c];

    float* __restrict__ season = out;
    float* __restrict__ trendo = out + (size_t)B_ * L_ * C_;

    for (int r = sub; r < TILE_L; r += 4) {
        const int l = l0 + r;

        float v[12];
#pragma unroll
        for (int i = 0; i < 12; ++i) {
            int gl = l + i - HALO_F;                         // edge-replication == clamp
            gl = gl < 0 ? 0 : (gl > (L_ - 1) ? (L_ - 1) : gl);
            v[i] = tile[(gl - row_start) * C_ + c];
        }

        const float s2  = v[5] + v[6];
        const float s4  = s2 + v[4] + v[7];
        const float s6  = s4 + v[3] + v[8];
        const float s12 = s6 + v[2] + v[1] + v[0] + v[9] + v[10] + v[11];

        const float xv = v[6];
        const float m0 = xv;
        const float m1 = s2  * 0.5f;
        const float m2 = s4  * 0.25f;
        const float m3 = s6  * (1.0f / 6.0f);
        const float m4 = s12 * (1.0f / 12.0f);

        const float g0l = fmaf(xv, w0, c0);
        const float g1l = fmaf(xv, w1, c1);
        const float g2l = fmaf(xv, w2, c2);
        const float g3l = fmaf(xv, w3, c3);
        const float g4l = fmaf(xv, w4, c4);
        const float mx  = fmaxf(fmaxf(fmaxf(g0l, g1l), fmaxf(g2l, g3l)), g4l);
        const float e0 = __expf(g0l - mx);
        const float e1 = __expf(g1l - mx);
        const float e2 = __expf(g2l - mx);
        const float e3 = __expf(g3l - mx);
        const float e4 = __expf(g4l - mx);
        const float den = e0 + e1 + e2 + e3 + e4;

        float tr = (m0 * e0 + m1 * e1 + m2 * e2 + m3 * e3 + m4 * e4) / den;
        tr = fmaf(tr, awc, abc);

        const size_t off = ((size_t)b * L_ + l) * C_ + c;    // coalesced across the wave
        __builtin_nontemporal_store(xv - tr, &season[off]);  // write-once: keep L2 for x
        __builtin_nontemporal_store(tr,      &trendo[off]);
    }
}

extern "C" void kernel_launch(void* const* d_in, const int* in_sizes, int n_in,
                              void* d_out, int out_size, void* d_ws, size_t ws_size,
                              hipStream_t stream) {
    const float* x  = (const float*)d_in[0];
    const float* mw = (const float*)d_in[1];
    const float* mb = (const float*)d_in[2];
    const float* aw = (const float*)d_in[3];
    const float* ab = (const float*)d_in[4];
    (void)in_sizes; (void)n_in; (void)out_size; (void)d_ws; (void)ws_size;

    const dim3 grid(B_ * (L_ / TILE_L));   // 4096 blocks, 8 wave32s each
    lma_tdm_kernel<<<grid, 256, 0, stream>>>(x, mw, mb, aw, ab, (float*)d_out);
}